// UnfoldDot_36971078484380
// MI455X (gfx1250) — compile-verified
//
#include <hip/hip_runtime.h>

typedef __attribute__((ext_vector_type(2))) float v2f;
typedef __attribute__((ext_vector_type(8))) float v8f;

#define BB 8
#define HH 16
#define TT 4096
#define DD 64
#define RR 127
#define WAVES_PER_BLOCK 8

// Compute one 16x16 (t,j) tile of QK^T via 16 chained V_WMMA_F32_16X16X4_F32
// and scatter it into the (t,r) band. GUARD=true only for the two edge
// j-tiles (jt==0: r may be <0; jt==8: r may be >=127); the 7 middle tiles
// are provably fully in-band and store branch-free.
//
// Out-of-range k rows: loads are clamped (always legal addresses); the
// zeroing is applied to the OUTPUT instead of the B operand, since C/D
// column N lives entirely in lanes with lane&15==N — the same lanes that
// own k row j0+N. c[v]*m (m in {0,1}) == zeroing the whole B column.
template <bool GUARD>
__device__ __forceinline__ void band_tile(
    const float* __restrict__ kbh, float* __restrict__ op,
    int jbase, int rbase0, int half, int h2, const v2f (&aq)[16])
{
    const unsigned jrow = (unsigned)jbase;              // negative -> huge
    const unsigned jcl  = jrow < (TT - 1u) ? jrow : (TT - 1u);
    const float    m    = (jrow < (unsigned)TT) ? 1.0f : 0.0f;

    const float* krow = kbh + (size_t)jcl * DD + h2;
    v2f bk[16];
#pragma unroll
    for (int i = 0; i < 16; ++i)
        bk[i] = *(const v2f*)(krow + 4 * i);            // b64, imm offsets

    v8f c = {};
#pragma unroll
    for (int i = 0; i < 16; ++i)
        c = __builtin_amdgcn_wmma_f32_16x16x4_f32(
                false, aq[i], false, bk[i], (short)0, c, false, false);

#pragma unroll
    for (int v = 0; v < 8; ++v) {
        const float val = c[v] * m;                     // zero OOB k columns
        if (GUARD) {
            const int r = rbase0 - (v + 8 * half);      // band coordinate
            if ((unsigned)r < (unsigned)RR)
                __builtin_nontemporal_store(val, op + v * 126);
        } else {
            __builtin_nontemporal_store(val, op + v * 126);
        }
    }
}

__global__ __launch_bounds__(32 * WAVES_PER_BLOCK)
void banded_qk_wmma_f32(const float* __restrict__ q,
                        const float* __restrict__ k,
                        float* __restrict__ out)
{
    const int lane    = threadIdx.x & 31;
    const int wave    = threadIdx.x >> 5;
    const int tileIdx = blockIdx.x * WAVES_PER_BLOCK + wave;  // 0..32767

    const int tt = tileIdx & (TT / 16 - 1);   // t-tile within (b,h)
    const int bh = tileIdx >> 8;              // 0..127
    const int t0 = tt * 16;

    const float* qbh = q   + (size_t)bh * TT * DD;
    const float* kbh = k   + (size_t)bh * TT * DD;
    float*       obh = out + (size_t)bh * TT * RR;

    // WMMA f32 16x16x4 lane roles (ISA 7.12.2, 32-bit layouts):
    //   A: row M = lane&15, VGPR v holds K = 2*(lane>>4)+v
    //   B: col N = lane&15, VGPR v holds K = 2*(lane>>4)+v
    //   C/D: col N = lane&15, VGPR v holds row M = v + 8*(lane>>4)
    const int N    = lane & 15;
    const int half = lane >> 4;
    const int h2   = half * 2;

    // Hoisted A (q) fragments: d = 4*i + h2 + {0,1}, 8B-aligned b64 loads.
    const float* qrow = qbh + (size_t)(t0 + N) * DD + h2;
    v2f aq[16];
#pragma unroll
    for (int i = 0; i < 16; ++i)
        aq[i] = *(const v2f*)(qrow + 4 * i);

    // Per-lane output base:
    //   addr(v, jt) = t0*127 + (v + 8*half)*126 + 16*jt + N - 1
    // so stores are op + 16*jt + v*126 with compile-time imm offsets v*504B.
    float* op0 = obh + ((long)t0 * RR + (8 * half) * 126 + N - 1);
    const int jb0 = t0 - 64 + N;              // this lane's k row at jt=0

    band_tile<true>(kbh, op0, jb0, N - 1, half, h2, aq);          // jt = 0
#pragma unroll 1
    for (int jt = 1; jt <= 7; ++jt)                               // in-band
        band_tile<false>(kbh, op0 + 16 * jt, jb0 + 16 * jt,
                         16 * jt - 1 + N, half, h2, aq);
    band_tile<true>(kbh, op0 + 128, jb0 + 128, 127 + N, half, h2, aq); // jt = 8
}

extern "C" void kernel_launch(void* const* d_in, const int* in_sizes, int n_in,
                              void* d_out, int out_size, void* d_ws, size_t ws_size,
                              hipStream_t stream) {
    const float* q = (const float*)d_in[0];
    const float* k = (const float*)d_in[1];
    // d_in[2] is `restrict` == 127, baked in as RR.
    float* out = (float*)d_out;

    const int total_tiles = BB * HH * (TT / 16);      // 32768 waves
    dim3 grid(total_tiles / WAVES_PER_BLOCK);         // 4096 blocks
    dim3 block(32 * WAVES_PER_BLOCK);                 // 8 waves / block
    hipLaunchKernelGGL(banded_qk_wmma_f32, grid, block, 0, stream, q, k, out);
}